// StickBreakingVAE_29231547416795
// MI455X (gfx1250) — compile-verified
//
#include <hip/hip_runtime.h>
#include <cstdint>
#include <cstddef>

// ---------------------------------------------------------------------------
// StickBreakingVAE reparametrize (Gauss_SBRK path), MI455X / gfx1250.
// Memory-bound streaming kernel: 512 MB total traffic -> ~22 us @ 23.3 TB/s.
// One wave32 per 256-wide row; async global->LDS double-buffered pipeline.
// ---------------------------------------------------------------------------

#define TOPICS     256
#define RPT        8                 // rows per tile (one per wave)
#define THREADS    256               // 8 waves of 32
#define TILE_ELEMS (RPT * TOPICS)    // 2048 floats = 8 KB per array
#define CLAMP_MIN  1e-4f
#define MAX_BLOCKS 2048

#define AS1 __attribute__((address_space(1)))
#define AS3 __attribute__((address_space(3)))

typedef float v4f __attribute__((ext_vector_type(4)));
typedef int   v4i __attribute__((ext_vector_type(4)));

#if __has_builtin(__builtin_amdgcn_global_load_async_to_lds_b128)
#define HAVE_ASYNC 1
#else
#define HAVE_ASYNC 0
#endif

#if HAVE_ASYNC
#define WAIT_ASYNC(n) asm volatile("s_wait_asynccnt " #n ::: "memory")
#else
#define WAIT_ASYNC(n) do {} while (0)
#endif

// Issue the copy of one 8-row tile (3 arrays x 8KB) into an LDS buffer.
// 256 threads x 16B per async op; 2 ops per array; 6 async ops per wave/tile.
__device__ __forceinline__ void issue_tile_copy(const float* __restrict__ g0,
                                                const float* __restrict__ g1,
                                                const float* __restrict__ g2,
                                                float* l0, float* l1, float* l2,
                                                int tid) {
  const float* gs[3] = {g0, g1, g2};
  float*       ls[3] = {l0, l1, l2};
#pragma unroll
  for (int a = 0; a < 3; ++a) {
#if HAVE_ASYNC
    __builtin_amdgcn_global_load_async_to_lds_b128(
        (AS1 v4i*)(gs[a] + tid * 4), (AS3 v4i*)(ls[a] + tid * 4), 0, 0);
    __builtin_amdgcn_global_load_async_to_lds_b128(
        (AS1 v4i*)(gs[a] + 1024 + tid * 4), (AS3 v4i*)(ls[a] + 1024 + tid * 4), 0, 0);
#else
    ((v4f*)(ls[a]))[tid]       = __builtin_nontemporal_load(((const v4f*)(gs[a])) + tid);
    ((v4f*)(ls[a]))[tid + 256] = __builtin_nontemporal_load(((const v4f*)(gs[a])) + tid + 256);
#endif
  }
}

__global__ __launch_bounds__(THREADS)
void sbvae_kernel(const float* __restrict__ mu, const float* __restrict__ sg,
                  const float* __restrict__ ep, float* __restrict__ out,
                  int ntiles) {
  __shared__ __align__(16) float buf[2][3][TILE_ELEMS];   // 48 KB

  const int tid  = threadIdx.x;
  const int wave = tid >> 5;
  const int lane = tid & 31;
  const int stride = gridDim.x;

  int tile = blockIdx.x;
  if (tile >= ntiles) return;

  // Prologue: prefetch tile 0 into buffer 0.
  {
    const size_t base = (size_t)tile * TILE_ELEMS;
    issue_tile_copy(mu + base, sg + base, ep + base,
                    buf[0][0], buf[0][1], buf[0][2], tid);
  }

  int b = 0;
  for (; tile < ntiles; tile += stride, b ^= 1) {
    const int nxt = tile + stride;
    if (nxt < ntiles) {
      const size_t nb = (size_t)nxt * TILE_ELEMS;
      issue_tile_copy(mu + nb, sg + nb, ep + nb,
                      buf[b ^ 1][0], buf[b ^ 1][1], buf[b ^ 1][2], tid);
      WAIT_ASYNC(6);   // current tile's 6 ops done; next tile's 6 in flight
    } else {
      WAIT_ASYNC(0);
    }
    __syncthreads();   // all waves' copies of tile `tile` visible

    // ---- compute: wave `wave` owns row `tile*8 + wave` ------------------
    const float* bm = buf[b][0];
    const float* bs = buf[b][1];
    const float* be = buf[b][2];
    const int lb = wave * TOPICS + lane * 8;   // 32B-aligned

    float m8[8], s8[8], e8[8];
    *(v4f*)&m8[0] = *(const v4f*)&bm[lb];
    *(v4f*)&m8[4] = *(const v4f*)&bm[lb + 4];
    *(v4f*)&s8[0] = *(const v4f*)&bs[lb];
    *(v4f*)&s8[4] = *(const v4f*)&bs[lb + 4];
    *(v4f*)&e8[0] = *(const v4f*)&be[lb];
    *(v4f*)&e8[4] = *(const v4f*)&be[lb + 4];

    // z = max(mu, eps_min) + sqrt(max(sigma, eps_min)) * noise
    float z[8];
    float part = 0.0f;
#pragma unroll
    for (int j = 0; j < 8; ++j) {
      const float zz = fmaxf(m8[j], CLAMP_MIN) +
                       __builtin_amdgcn_sqrtf(fmaxf(s8[j], CLAMP_MIN)) * e8[j];
      z[j] = zz;
      part += zz;
    }

    // wave32 sum reduction (5 steps)
#pragma unroll
    for (int o = 16; o > 0; o >>= 1) part += __shfl_xor(part, o, 32);

    // fast reciprocal + one Newton refinement
    float r = __builtin_amdgcn_rcpf(part);
    r = r * (2.0f - part * r);
#pragma unroll
    for (int j = 0; j < 8; ++j) z[j] *= r;

    // per-lane inclusive products of (1 - z)
    float p[8];
    float acc = 1.0f;
#pragma unroll
    for (int j = 0; j < 8; ++j) {
      acc *= (1.0f - z[j]);
      p[j] = acc;
    }

    // multiplicative exclusive scan of lane totals across the wave (5 steps)
    float scan = p[7];
#pragma unroll
    for (int o = 1; o < 32; o <<= 1) {
      const float up = __shfl_up(scan, o, 32);
      if (lane >= o) scan *= up;
    }
    float excl = __shfl_up(scan, 1, 32);
    if (lane == 0) excl = 1.0f;

    // pi[k] = z[k] * prod_{j<k} (1 - z[j])
    float o8[8];
    o8[0] = z[0] * excl;
#pragma unroll
    for (int j = 1; j < 8; ++j) o8[j] = z[j] * excl * p[j - 1];

    float* op = out + (size_t)(tile * RPT + wave) * TOPICS + lane * 8;
    __builtin_nontemporal_store(*(v4f*)&o8[0], (v4f*)op);
    __builtin_nontemporal_store(*(v4f*)&o8[4], (v4f*)(op + 4));

    __syncthreads();   // buffer b fully consumed; safe to refill next round
  }
}

extern "C" void kernel_launch(void* const* d_in, const int* in_sizes, int n_in,
                              void* d_out, int out_size, void* d_ws, size_t ws_size,
                              hipStream_t stream) {
  const float* mu = (const float*)d_in[0];
  const float* sg = (const float*)d_in[1];
  const float* ep = (const float*)d_in[2];
  float* out = (float*)d_out;

  const int rows   = in_sizes[0] / TOPICS;   // 131072
  const int ntiles = rows / RPT;             // 16384
  const int grid   = (ntiles < MAX_BLOCKS) ? ntiles : MAX_BLOCKS;

  sbvae_kernel<<<grid, THREADS, 0, stream>>>(mu, sg, ep, out, ntiles);
}